// StackedDynamicGraphNeuralNetwork_10204842295628
// MI455X (gfx1250) — compile-verified
//
#include <hip/hip_runtime.h>

// Problem constants (from reference setup_inputs)
#define BATCH 16
#define NSEQ  2048
#define DDIM  128
#define NUM_LAYERS 3
#define KSPLIT 8            // split-K factor for M = X^T (r ⊙ h)

typedef __attribute__((ext_vector_type(2))) float v2f;
typedef __attribute__((ext_vector_type(8))) float v8f;

// ---------------------------------------------------------------------------
// Kernel 1: per-batch column sum  t[b][d] = sum_n x[b][n][d]
// ---------------------------------------------------------------------------
__global__ void k_colsum(const float* __restrict__ x, float* __restrict__ t) {
    const int b = blockIdx.x;
    const int d = threadIdx.x;
    const float* xb = x + (size_t)b * NSEQ * DDIM;
    float s = 0.0f;
#pragma unroll 8
    for (int n = 0; n < NSEQ; ++n) s += xb[(size_t)n * DDIM + d];
    t[b * DDIM + d] = s;
}

// ---------------------------------------------------------------------------
// Kernel 2: rinv[b][i] = rsqrt(x_i . t_b + 1)   (rowsum of S = X X^T + I)
// one wave32 per row; shuffle reduction.
// ---------------------------------------------------------------------------
__global__ void k_rinv(const float* __restrict__ x, const float* __restrict__ t,
                       float* __restrict__ rinv) {
    const int gw   = (int)((blockIdx.x * blockDim.x + threadIdx.x) >> 5);
    const int lane = threadIdx.x & 31;
    const int b = gw >> 11;          // NSEQ = 2048 rows per batch
    const int i = gw & (NSEQ - 1);
    const float* xr = x + ((size_t)b * NSEQ + i) * DDIM;
    const float* tb = t + b * DDIM;
    float s = 0.0f;
#pragma unroll
    for (int k = lane; k < DDIM; k += 32) s += xr[k] * tb[k];
#pragma unroll
    for (int off = 16; off > 0; off >>= 1) s += __shfl_xor(s, off, 32);
    s += 1.0f;                        // self-loop diagonal of (S + I) rowsum
    float r = (s == 0.0f) ? 1.0f : s;
    if (lane == 0) rinv[b * NSEQ + i] = rsqrtf(r);
}

// ---------------------------------------------------------------------------
// V_WMMA_F32_16X16X4_F32 operand mapping (wave32):
//   A 16x4:  lane(0..15)=row m, half=lane>>4 -> k0=2*half; v2f = A[m][k0..k0+1]
//   B 4x16:  lane(0..15)=col n;               v2f = B[k0][n], B[k0+1][n]
//   C/D:     vgpr r -> row (r + 8*half), col = lane & 15
// ---------------------------------------------------------------------------

// Kernel 3: h = X W + bias.  One wave per 16x64 strip: A pair loaded once,
// reused by 4 independent WMMAs per k-step. Grid sized exactly (EXEC all 1s).
__global__ void k_gemm_h(const float* __restrict__ x, const float* __restrict__ Wm,
                         const float* __restrict__ bias, float* __restrict__ h) {
    const int gw   = (int)((blockIdx.x * blockDim.x + threadIdx.x) >> 5);
    const int lane = threadIdx.x & 31;
    const int lm = lane & 15, half = lane >> 4, ko = half * 2;
    const int mt = gw >> 1;           // 16-row strip index
    const int ng = gw & 1;            // which 64-col half
    const int row0 = mt * 16, colb = ng * 64;
    const float* A = x + ((size_t)row0 + lm) * DDIM;
    v8f c[4] = {};
#pragma unroll 4
    for (int k = 0; k < DDIM; k += 4) {
        const v2f a = *(const v2f*)(A + k + ko);          // packed, reused x4
#pragma unroll
        for (int j = 0; j < 4; ++j) {
            v2f bv;
            bv.x = Wm[(size_t)(k + ko) * DDIM + colb + j * 16 + lm];
            bv.y = Wm[(size_t)(k + ko + 1) * DDIM + colb + j * 16 + lm];
            c[j] = __builtin_amdgcn_wmma_f32_16x16x4_f32(false, a, false, bv,
                                                         (short)0, c[j], false, false);
        }
    }
#pragma unroll
    for (int j = 0; j < 4; ++j) {
        const float bb = bias[colb + j * 16 + lm];
#pragma unroll
        for (int r = 0; r < 8; ++r)
            h[((size_t)row0 + r + 8 * half) * DDIM + colb + j * 16 + lm] = c[j][r] + bb;
    }
}

// Kernel 4: split-K partials of Mmat_b = X_b^T (rinv_b ⊙ h_b).
// 8192 waves: one 16x16 tile x one K-chunk of 256 rows each; deterministic
// (each wave owns a distinct partial-buffer slot; no float atomics).
__global__ void k_gemm_m_part(const float* __restrict__ x, const float* __restrict__ h,
                              const float* __restrict__ rinv, float* __restrict__ part) {
    const int gw   = (int)((blockIdx.x * blockDim.x + threadIdx.x) >> 5);
    const int lane = threadIdx.x & 31;
    const int lm = lane & 15, half = lane >> 4, ko = half * 2;
    const int b    = gw >> 9;         // 512 waves per batch = 8 chunks * 64 tiles
    const int rem  = gw & 511;
    const int kc   = rem >> 6;
    const int tile = rem & 63;
    const int mt = tile >> 3, nt = tile & 7;
    const int m0 = mt * 16, n0 = nt * 16;
    const int kbeg = kc * (NSEQ / KSPLIT);
    const float* X = x + (size_t)b * NSEQ * DDIM;
    const float* H = h + (size_t)b * NSEQ * DDIM;
    const float* R = rinv + b * NSEQ;
    v8f c = {};
#pragma unroll 4
    for (int k = kbeg; k < kbeg + NSEQ / KSPLIT; k += 4) {
        const int kk = k + ko;
        v2f a, bv;
        a.x = X[(size_t)kk * DDIM + m0 + lm];                 // A = X^T
        a.y = X[(size_t)(kk + 1) * DDIM + m0 + lm];
        bv.x = R[kk]     * H[(size_t)kk * DDIM + n0 + lm];    // B = rinv ⊙ h
        bv.y = R[kk + 1] * H[(size_t)(kk + 1) * DDIM + n0 + lm];
        c = __builtin_amdgcn_wmma_f32_16x16x4_f32(false, a, false, bv,
                                                  (short)0, c, false, false);
    }
    float* P = part + ((size_t)kc * BATCH + b) * DDIM * DDIM;
#pragma unroll
    for (int r = 0; r < 8; ++r)
        P[(size_t)(m0 + r + 8 * half) * DDIM + n0 + lm] = c[r];
}

// Kernel 5: deterministic 8-way reduction of the split-K partials.
__global__ void k_reduce_m(const float* __restrict__ part, float* __restrict__ Mmat) {
    const size_t i = (size_t)blockIdx.x * blockDim.x + threadIdx.x;
    float s = 0.0f;
#pragma unroll
    for (int kc = 0; kc < KSPLIT; ++kc)
        s += part[(size_t)kc * BATCH * DDIM * DDIM + i];
    Mmat[i] = s;
}

// Kernel 6: out = relu( rinv ⊙ (X Mmat_b) + rinv^2 ⊙ h ).  16x64 per wave.
__global__ void k_gemm_out(const float* __restrict__ x, const float* __restrict__ Mmat,
                           const float* __restrict__ h, const float* __restrict__ rinv,
                           float* __restrict__ out) {
    const int gw   = (int)((blockIdx.x * blockDim.x + threadIdx.x) >> 5);
    const int lane = threadIdx.x & 31;
    const int lm = lane & 15, half = lane >> 4, ko = half * 2;
    const int b    = gw >> 8;         // (NSEQ/16)*(DDIM/64) = 256 waves per batch
    const int tile = gw & 255;
    const int mt = tile >> 1;
    const int ng = tile & 1;
    const int row0 = mt * 16, colb = ng * 64;
    const float* A  = x + (size_t)b * NSEQ * DDIM + ((size_t)row0 + lm) * DDIM;
    const float* Bm = Mmat + (size_t)b * DDIM * DDIM;
    const float* H  = h + (size_t)b * NSEQ * DDIM;
    const float* R  = rinv + b * NSEQ;
    v8f c[4] = {};
#pragma unroll 4
    for (int k = 0; k < DDIM; k += 4) {
        const v2f a = *(const v2f*)(A + k + ko);
#pragma unroll
        for (int j = 0; j < 4; ++j) {
            v2f bv;
            bv.x = Bm[(size_t)(k + ko) * DDIM + colb + j * 16 + lm];
            bv.y = Bm[(size_t)(k + ko + 1) * DDIM + colb + j * 16 + lm];
            c[j] = __builtin_amdgcn_wmma_f32_16x16x4_f32(false, a, false, bv,
                                                         (short)0, c[j], false, false);
        }
    }
#pragma unroll
    for (int r = 0; r < 8; ++r) {
        const int row = row0 + r + 8 * half;
        const float ri = R[row];
        const float ri2 = ri * ri;
#pragma unroll
        for (int j = 0; j < 4; ++j) {
            const float hv = H[(size_t)row * DDIM + colb + j * 16 + lm];
            const float v  = ri * c[j][r] + ri2 * hv;
            out[(size_t)b * NSEQ * DDIM + (size_t)row * DDIM + colb + j * 16 + lm] =
                fmaxf(v, 0.0f);
        }
    }
}

// ---------------------------------------------------------------------------
extern "C" void kernel_launch(void* const* d_in, const int* in_sizes, int n_in,
                              void* d_out, int out_size, void* d_ws, size_t ws_size,
                              hipStream_t stream) {
    (void)in_sizes; (void)n_in; (void)out_size; (void)ws_size;
    const float* x_in = (const float*)d_in[0];
    const float* Ws   = (const float*)d_in[1];   // [3, 128, 128]
    const float* bs   = (const float*)d_in[2];   // [3, 128]
    float* out = (float*)d_out;

    const size_t XN = (size_t)BATCH * NSEQ * DDIM;       // 4,194,304 floats
    const size_t MM = (size_t)BATCH * DDIM * DDIM;       //   262,144 floats
    float* ws = (float*)d_ws;
    float* buf0 = ws;                        // layer ping buffer     (16 MB)
    float* buf1 = buf0 + XN;                 // layer pong buffer     (16 MB)
    float* hbuf = buf1 + XN;                 // h = XW + b            (16 MB)
    float* pbuf = hbuf + XN;                 // split-K partials       (8 MB)
    float* mbuf = pbuf + (size_t)KSPLIT * MM;// Mmat                   (1 MB)
    float* tbuf = mbuf + MM;                 // column sums [16,128]
    float* rbuf = tbuf + BATCH * DDIM;       // rinv [16,2048]
    // total ≈ 57 MB of workspace

    const int blocks_big  = (BATCH * NSEQ / 16) * (DDIM / 64) * 32 / 256; // 512
    const int blocks_mp   = BATCH * KSPLIT * 64 * 32 / 256;               // 1024
    const int blocks_red  = (int)(MM / 256);                              // 1024
    const int blocks_rinv = BATCH * NSEQ * 32 / 256;                      // 4096

    for (int l = 0; l < NUM_LAYERS; ++l) {
        const float* in = (l == 0) ? x_in : ((l == 1) ? buf0 : buf1);
        float* o = (l == NUM_LAYERS - 1) ? out : ((l == 0) ? buf0 : buf1);
        const float* W = Ws + (size_t)l * DDIM * DDIM;
        const float* bias = bs + (size_t)l * DDIM;

        k_colsum<<<BATCH, DDIM, 0, stream>>>(in, tbuf);
        k_rinv<<<blocks_rinv, 256, 0, stream>>>(in, tbuf, rbuf);
        k_gemm_h<<<blocks_big, 256, 0, stream>>>(in, W, bias, hbuf);
        k_gemm_m_part<<<blocks_mp, 256, 0, stream>>>(in, hbuf, rbuf, pbuf);
        k_reduce_m<<<blocks_red, 256, 0, stream>>>(pbuf, mbuf);
        k_gemm_out<<<blocks_big, 256, 0, stream>>>(in, mbuf, hbuf, rbuf, o);
    }
}